// PolynomialModel_3204045603237
// MI455X (gfx1250) — compile-verified
//
#include <hip/hip_runtime.h>

typedef __attribute__((ext_vector_type(2))) float v2f;
typedef __attribute__((ext_vector_type(4))) float v4f;
typedef __attribute__((ext_vector_type(8))) float v8f;

#define DIMS 64
#define ROWS_PER_BLOCK 128
#define XSTRIDE 68    // 68 % 64 = 4 -> 16 rows of a column read hit 16 distinct banks
#define BSTRIDE 130   // inner row of Bsw in float2 units (128 used + 2 pad)

__global__ __launch_bounds__(256)
void poly_wmma_kernel(const float* __restrict__ X,
                      const float* __restrict__ W,
                      const float* __restrict__ Bv,
                      float* __restrict__ out)
{
    __shared__ float Xs[ROWS_PER_BLOCK][XSTRIDE]; // 34,816 B : block's X tile
    __shared__ v2f   Bsw[16][BSTRIDE];            // 16,640 B : B fragments, pre-swizzled
    __shared__ float Wsh[DIMS];                   //    256 B : linear weights

    const int tid     = threadIdx.x;
    const int rowbase = blockIdx.x * ROWS_PER_BLOCK;

    if (tid < DIMS) Wsh[tid] = W[1 + tid];

    // Unpack upper-triangular quadratic weights directly into the WMMA B-fragment
    // layout.  Fragment for (ks, half) at column n needs
    //   { U^T[4ks+2half][n], U^T[4ks+2half+1][n] }  where U^T[k][n] = Q(n,k) if k>=n.
    // sklearn triu order: pair (i,j), i<=j  ->  W[65 + i*64 - i(i-1)/2 + (j-i)]
    for (int e = tid; e < 16 * 128; e += 256) {
        int ks   = e >> 7;         // k-step 0..15
        int t    = e & 127;
        int n    = t >> 1;         // column 0..63
        int half = t & 1;          // lane half
        int k    = 4 * ks + 2 * half;
        v2f v;
        v.x = (k     >= n) ? W[65 + n * DIMS - (n * (n - 1)) / 2 + (k     - n)] : 0.0f;
        v.y = (k + 1 >= n) ? W[65 + n * DIMS - (n * (n - 1)) / 2 + (k + 1 - n)] : 0.0f;
        Bsw[ks][2 * n + half] = v;
    }

    // Stage 128x64 X tile, coalesced float4 loads (stream X exactly once).
    {
        const v4f* Xg = (const v4f*)(X + (size_t)rowbase * DIMS);
        #pragma unroll
        for (int i = 0; i < 8; ++i) {
            int g = tid + i * 256;        // float4 index in tile (0..2047)
            int r = g >> 4;               // row (16 float4 per row)
            int c = (g & 15) << 2;        // col
            v4f val = Xg[g];
            *(v4f*)&Xs[r][c] = val;
        }
    }
    __syncthreads();

    const int wave = tid >> 5;
    const int lane = tid & 31;
    const int half = lane >> 4;   // 0 -> lanes 0-15, 1 -> lanes 16-31
    const int l15  = lane & 15;
    const int wr   = wave * 16;   // this wave's 16-row tile inside the block

    // A fragments (16x4 f32, 2 VGPRs): lane = row M, VGPR pair = K {k0+2h, k0+2h+1}
    v2f afrag[16];
    #pragma unroll
    for (int ks = 0; ks < 16; ++ks) {
        int k = ks * 4 + half * 2;
        afrag[ks].x = Xs[wr + l15][k];
        afrag[ks].y = Xs[wr + l15][k + 1];
    }

    v8f acc[4];
    #pragma unroll
    for (int nt = 0; nt < 4; ++nt) acc[nt] = (v8f)(0.0f);

    // Y = X * U^T : 64 v_wmma_f32_16x16x4_f32 per wave; each B fragment is a
    // single conflict-free ds_load_b64 (32 lanes -> one contiguous 256B window).
    #pragma unroll
    for (int nt = 0; nt < 4; ++nt) {
        #pragma unroll
        for (int ks = 0; ks < 16; ++ks) {
            v2f bfrag = Bsw[ks][2 * (nt * 16 + l15) + half];
            acc[nt] = __builtin_amdgcn_wmma_f32_16x16x4_f32(
                false, afrag[ks], false, bfrag, (short)0, acc[nt], false, false);
        }
    }

    const float bias = W[0] + Bv[0];

    // out[m] = bias + sum_n X[m][n] * (w[n] + Y[m][n])
    // C/D layout: element r of acc holds M = r + 8*half, N = l15 + 16*nt
    #pragma unroll
    for (int r = 0; r < 8; ++r) {
        int m = r + 8 * half;
        float s = 0.0f;
        #pragma unroll
        for (int nt = 0; nt < 4; ++nt) {
            int n = nt * 16 + l15;
            s += Xs[wr + m][n] * (Wsh[n] + acc[nt][r]);
        }
        // reduce over the 16 lanes holding the same output row
        s += __shfl_xor(s, 1, 16);
        s += __shfl_xor(s, 2, 16);
        s += __shfl_xor(s, 4, 16);
        s += __shfl_xor(s, 8, 16);
        if (l15 == 0) out[rowbase + wr + m] = bias + s;
    }
}

extern "C" void kernel_launch(void* const* d_in, const int* in_sizes, int n_in,
                              void* d_out, int out_size, void* d_ws, size_t ws_size,
                              hipStream_t stream) {
    const float* x  = (const float*)d_in[0];   // [32768, 64]
    const float* W  = (const float*)d_in[1];   // [1, 2145]
    const float* bv = (const float*)d_in[2];   // [1]
    float* out = (float*)d_out;                // [32768]

    const int batch = in_sizes[0] / DIMS;              // 32768
    const int grid  = batch / ROWS_PER_BLOCK;          // 256 blocks of 256 threads
    poly_wmma_kernel<<<grid, 256, 0, stream>>>(x, W, bv, out);
}